// TopologyTypePredictor_76699525972070
// MI455X (gfx1250) — compile-verified
//
#include <hip/hip_runtime.h>
#include <hip/hip_bf16.h>

typedef __attribute__((ext_vector_type(2))) float v2f;
typedef __attribute__((ext_vector_type(8))) float v8f;

#define D_MODEL 256
#define NTYPES 7
#define BATCH 2
#define N_LANES 512
#define RB 8             // phase-1 rows per block (register blocking)
#define TI 16            // phase-2 i rows per block
#define TJ 32            // phase-2 j rows per block (two 16-row M-groups)
#define ROW_STRIDE 260   // 256 + 4 pad floats -> bank-conflict-free ds_load_b64
// dynamic LDS: hjS[TJ*260] | hiS[TI*260] | w2S[64*64]
#define SMEM_FLOATS ((TI + TJ) * ROW_STRIDE + 64 * 64)

// ---------------- Phase 1: hi = X@W1a + b1 ; hj = X@W1b ----------------
// Register-blocked over 8 rows: each thread owns one output column of 8 hi
// rows + 8 hj rows. W1 elements are loaded ONCE per block (8x less L2
// traffic than row-per-block), X reads are block-uniform -> scalar loads
// through the constant cache. 16 FMA per 2 vector loads.
__global__ __launch_bounds__(256) void phase1_gemm(
    const float* __restrict__ X, const float* __restrict__ W1,
    const float* __restrict__ b1, float* __restrict__ hi,
    float* __restrict__ hj) {
  const int row0 = blockIdx.x * RB;  // 128 blocks over 1024 rows
  const int t = threadIdx.x;         // output column 0..255

  float acc_i[RB], acc_j[RB];
  const float bfold = b1[t];  // fold b1 into hi so phase 2 only adds hi+hj
#pragma unroll
  for (int r = 0; r < RB; ++r) {
    acc_i[r] = bfold;
    acc_j[r] = 0.f;
  }
#pragma unroll 4
  for (int d = 0; d < D_MODEL; ++d) {
    const float wa = W1[d * D_MODEL + t];
    const float wb = W1[(D_MODEL + d) * D_MODEL + t];
#pragma unroll
    for (int r = 0; r < RB; ++r) {
      const float x = X[(row0 + r) * D_MODEL + d];  // uniform -> s_load
      acc_i[r] = fmaf(x, wa, acc_i[r]);
      acc_j[r] = fmaf(x, wb, acc_j[r]);
    }
  }
#pragma unroll
  for (int r = 0; r < RB; ++r) {
    hi[(row0 + r) * D_MODEL + t] = acc_i[r];
    hj[(row0 + r) * D_MODEL + t] = acc_j[r];
  }
}

// ---------------- Phase 2: logits = relu(hi[i]+hj[j]) @ W2 + b2 --------
// Block = 16 i x 32 j. Wave w owns j-group (w>>2) and 4 i rows -> 4
// accumulators. K-chunk loop outer: the hj fragment and the W2 B-fragment
// are loaded ONCE per chunk and reused by all 4 WMMAs (1.5 ds_load_b64 and
// 2 VALU per v_wmma_f32_16x16x4_f32).
__global__ __launch_bounds__(256) void phase2_pair_logits(
    const float* __restrict__ hi, const float* __restrict__ hj,
    const float* __restrict__ W2, const float* __restrict__ b2,
    float* __restrict__ out) {
  extern __shared__ float smem[];
  float* hjS = smem;                    // TJ * ROW_STRIDE
  float* hiS = smem + TJ * ROW_STRIDE;  // TI * ROW_STRIDE
  float* w2S = smem + (TI + TJ) * ROW_STRIDE;  // [chunk][lane][2]

  const int blk = blockIdx.x;  // 2 * 32 * 16 = 1024 blocks
  const int bb = blk >> 9;
  const int rem = blk & 511;
  const int i0 = (rem >> 4) * TI;  // 32 i-tiles
  const int j0 = (rem & 15) * TJ;  // 16 j-tiles
  const int tid = threadIdx.x;
  const int wave = tid >> 5;
  const int lane = tid & 31;
  const int sub = lane >> 4;  // half-wave: K sub-pair select
  const int m = lane & 15;    // A row within M-group / D column

  const float* hiB = hi + (bb * N_LANES + i0) * D_MODEL;
  const float* hjB = hj + (bb * N_LANES + j0) * D_MODEL;

  // Stage hi/hj tiles into padded LDS rows (float4; ROW_STRIDE*4 = 1040 is a
  // multiple of 16 so LDS stores stay 16B-aligned).
  for (int e = tid; e < TI * (D_MODEL / 4); e += 256) {
    int r = e >> 6, c4 = e & 63;
    float4 a = ((const float4*)hiB)[r * (D_MODEL / 4) + c4];
    *(float4*)&hiS[r * ROW_STRIDE + c4 * 4] = a;
  }
  for (int e = tid; e < TJ * (D_MODEL / 4); e += 256) {
    int r = e >> 6, c4 = e & 63;
    float4 a = ((const float4*)hjB)[r * (D_MODEL / 4) + c4];
    *(float4*)&hjS[r * ROW_STRIDE + c4 * 4] = a;
  }
  // Stage W2 pre-swizzled in the exact 4x16 f32 B-operand layout:
  // lane L, chunk c holds {W2p[4c+2*(L/16)+0][L%16], W2p[...+1][L%16]}.
  for (int e = tid; e < 64 * 64; e += 256) {
    int c = e >> 6, le = (e >> 1) & 31, which = e & 1;
    int k = 4 * c + 2 * (le >> 4) + which;
    int n = le & 15;
    w2S[e] = (n < NTYPES) ? W2[k * NTYPES + n] : 0.f;
  }
  __syncthreads();

  const float bv = (m < NTYPES) ? b2[m] : 0.f;  // b2 folded into C init
  const int jg = wave >> 2;        // j-group for this wave
  const int ilb = (wave & 3) * 4;  // first of 4 i rows for this wave
  const float* hjR = &hjS[(jg * 16 + m) * ROW_STRIDE];
  const float* hiR = &hiS[ilb * ROW_STRIDE];

  v8f acc0 = {bv, bv, bv, bv, bv, bv, bv, bv};
  v8f acc1 = acc0, acc2 = acc0, acc3 = acc0;

#pragma unroll 4
  for (int c = 0; c < 64; ++c) {
    const int k = 4 * c + 2 * sub;
    v2f hjv = *(const v2f*)&hjR[k];                   // shared by 4 WMMAs
    v2f bmat = *(const v2f*)&w2S[c * 64 + lane * 2];  // shared by 4 WMMAs
    v2f h0 = *(const v2f*)&hiR[k] + hjv;
    v2f h1 = *(const v2f*)&hiR[ROW_STRIDE + k] + hjv;
    v2f h2 = *(const v2f*)&hiR[2 * ROW_STRIDE + k] + hjv;
    v2f h3 = *(const v2f*)&hiR[3 * ROW_STRIDE + k] + hjv;
    h0.x = fmaxf(h0.x, 0.f); h0.y = fmaxf(h0.y, 0.f);
    h1.x = fmaxf(h1.x, 0.f); h1.y = fmaxf(h1.y, 0.f);
    h2.x = fmaxf(h2.x, 0.f); h2.y = fmaxf(h2.y, 0.f);
    h3.x = fmaxf(h3.x, 0.f); h3.y = fmaxf(h3.y, 0.f);
    acc0 = __builtin_amdgcn_wmma_f32_16x16x4_f32(false, h0, false, bmat,
                                                 (short)0, acc0, false, false);
    acc1 = __builtin_amdgcn_wmma_f32_16x16x4_f32(false, h1, false, bmat,
                                                 (short)0, acc1, false, false);
    acc2 = __builtin_amdgcn_wmma_f32_16x16x4_f32(false, h2, false, bmat,
                                                 (short)0, acc2, false, false);
    acc3 = __builtin_amdgcn_wmma_f32_16x16x4_f32(false, h3, false, bmat,
                                                 (short)0, acc3, false, false);
  }

  // Store: D layout -> lane L, VGPR v holds (M = v + 8*sub, N = L%16).
  const int jcol = j0 + jg * 16 + 8 * sub;
  v8f accs[4] = {acc0, acc1, acc2, acc3};
#pragma unroll
  for (int t = 0; t < 4; ++t) {
    const int gi = i0 + ilb + t;
    float* o = out + ((size_t)(bb * N_LANES + gi) * N_LANES + jcol) * NTYPES;
#pragma unroll
    for (int v = 0; v < 8; ++v) {
      if (m < NTYPES) o[(size_t)v * NTYPES + m] = accs[t][v];
    }
  }
}

extern "C" void kernel_launch(void* const* d_in, const int* in_sizes, int n_in,
                              void* d_out, int out_size, void* d_ws,
                              size_t ws_size, hipStream_t stream) {
  const float* X = (const float*)d_in[0];   // (2,512,256)
  const float* W1 = (const float*)d_in[1];  // (512,256)
  const float* b1 = (const float*)d_in[2];  // (256,)
  const float* W2 = (const float*)d_in[3];  // (256,7)
  const float* b2 = (const float*)d_in[4];  // (7,)
  float* out = (float*)d_out;               // (2,512,512,7)

  float* hi = (float*)d_ws;  // 2*512*256 floats
  float* hj = hi + BATCH * N_LANES * D_MODEL;

  phase1_gemm<<<(BATCH * N_LANES) / RB, 256, 0, stream>>>(X, W1, b1, hi, hj);

  const int blocks = BATCH * (N_LANES / TI) * (N_LANES / TJ);  // 1024
  phase2_pair_logits<<<blocks, 256, SMEM_FLOATS * sizeof(float), stream>>>(
      hi, hj, W2, b2, out);
}